// Net_38405597561668
// MI455X (gfx1250) — compile-verified
//
#include <hip/hip_runtime.h>
#include <math.h>

#define TWO_PI 6.283185307179586f
#define EPS    1e-4f

typedef __attribute__((ext_vector_type(2))) float v2f;
typedef __attribute__((ext_vector_type(8))) float v8f;

__device__ __forceinline__ float det2(const float* g) {
  // gaussian layout: [w, px, py, c00, c01, c10, c11]
  return g[3] * g[6] - g[4] * g[5];
}

// ---------------------------------------------------------------------------
// Stage 1: per-gaussian batchnorm on in_x (4,1,128,7)
// scale[g] = mean_b |w| * 2pi * sqrt(det C); w' = w / (scale + eps)
// ---------------------------------------------------------------------------
__global__ void bn1_kernel(const float* __restrict__ in, float* __restrict__ out) {
  int g = threadIdx.x;              // 128 threads
  float s = 0.f;
  for (int b = 0; b < 4; ++b) {
    const float* m = in + (size_t)(b * 128 + g) * 7;
    s += fabsf(m[0]) * TWO_PI * sqrtf(det2(m));
  }
  s *= 0.25f;
  for (int b = 0; b < 4; ++b) {
    const float* m = in + (size_t)(b * 128 + g) * 7;
    float* o = out + (size_t)(b * 128 + g) * 7;
    o[0] = m[0] / (s + EPS);
    #pragma unroll
    for (int q = 1; q < 7; ++q) o[q] = m[q];
  }
}

// ---------------------------------------------------------------------------
// Convolve: (B,Li,Ni,7) x (Lo,Li,K,7) -> (B,Lo,Li*Ni*K,7)
// flat j = (li*Ni + ni)*K + k
// ---------------------------------------------------------------------------
__global__ void conv_kernel(const float* __restrict__ in, const float* __restrict__ ker,
                            float* __restrict__ out,
                            int B, int Lo, int Li, int Ni, int K) {
  int idx = blockIdx.x * blockDim.x + threadIdx.x;
  int N = Li * Ni * K;
  int total = B * Lo * N;
  if (idx >= total) return;
  int j  = idx % N;
  int t  = idx / N;
  int lo = t % Lo;
  int b  = t / Lo;
  int k  = j % K;
  int s2 = j / K;
  int ni = s2 % Ni;
  int li = s2 / Ni;
  const float* gi = in  + (size_t)((b * Li + li) * Ni + ni) * 7;
  const float* gk = ker + (size_t)((lo * Li + li) * K + k) * 7;
  float di = det2(gi), dk = det2(gk);
  float c3 = gi[3] + gk[3], c4 = gi[4] + gk[4];
  float c5 = gi[5] + gk[5], c6 = gi[6] + gk[6];
  float ds = c3 * c6 - c4 * c5;
  float norm = TWO_PI * sqrtf(di * dk / ds);
  float* o = out + (size_t)((b * Lo + lo) * N + j) * 7;
  o[0] = gi[0] * gk[0] * norm;
  o[1] = gi[1] + gk[1];
  o[2] = gi[2] + gk[2];
  o[3] = c3; o[4] = c4; o[5] = c5; o[6] = c6;
}

// ---------------------------------------------------------------------------
// bias_relu: per (b,l) all-pairs evaluation via rank-6 bilinear WMMA,
// then bias/ReLU and deterministic top-KSEL selection.
// One workgroup (256 thr = 8 waves) per (b,l).
//
// q_ij = a_i . b_j with a = [x^2, y^2, xy, x, y, 1, 0, 0] (K padded 6->8).
// b-vectors stored in LDS with stride 8 so both half-groups of the wave
// fetch their (K0,K1)/(K2,K3) and (K4,K5)/(pad,pad) pairs with branchless
// ds_load_b64 at j*8 + half*2 and j*8 + 4 + half*2. The padded slots are
// set to zero; the matching A entries (a_hi of lanes 16-31) are zero.
// ---------------------------------------------------------------------------
template<int N, int KSEL>
__global__ void __launch_bounds__(256)
bias_relu_kernel(const float* __restrict__ m, const float* __restrict__ bias,
                 float* __restrict__ out, int L) {
  __shared__ alignas(16) float sB[N * 8]; // padded b-vectors
  __shared__ float sW[N];                 // weights
  __shared__ float sVals[N];              // row sums
  __shared__ float redV[256];
  __shared__ int   redI[256];

  const int tid  = threadIdx.x;
  const int l    = blockIdx.x % L;
  const int b    = blockIdx.x / L;
  const float* mb = m + (size_t)(b * L + l) * N * 7;

  // ---- Phase 0: build padded b-vectors in LDS ----
  for (int j = tid; j < N; j += 256) {
    const float* g = mb + (size_t)j * 7;
    float px = g[1], py = g[2];
    float c00 = g[3], c01 = g[4], c10 = g[5], c11 = g[6];
    float inv = 1.0f / (c00 * c11 - c01 * c10);
    float i00 = c11 * inv, i01 = -c01 * inv, i11 = c00 * inv;
    float* bvec = sB + j * 8;
    bvec[0] = i00;
    bvec[1] = i11;
    bvec[2] = 2.0f * i01;
    bvec[3] = -2.0f * (i00 * px + i01 * py);
    bvec[4] = -2.0f * (i01 * px + i11 * py);
    bvec[5] = i00 * px * px + 2.0f * i01 * px * py + i11 * py * py;
    bvec[6] = 0.0f;
    bvec[7] = 0.0f;
    sW[j] = g[0];
  }
  __syncthreads();

  // ---- Phase 1: WMMA tiles: two chained 16x16x4 f32 WMMAs per tile ----
  const int wid  = tid >> 5;
  const int lane = tid & 31;
  const int half = lane >> 4;     // K-split half (A/B layout)
  const int c    = lane & 15;     // row (A) / col (B) index inside tile
  const int nTiles = N / 16;

  for (int rt = wid; rt < nTiles; rt += 8) {
    const int r0  = rt * 16;
    const int row = r0 + c;
    float px = mb[(size_t)row * 7 + 1];
    float py = mb[(size_t)row * 7 + 2];
    // branchless per-lane A fragment (ternaries -> v_cndmask, no EXEC diamonds)
    v2f a_lo, a_hi;
    a_lo.x = half ? px * py : px * px;
    a_lo.y = half ? px      : py * py;
    a_hi.x = half ? 0.0f    : py;
    a_hi.y = half ? 0.0f    : 1.0f;

    v8f acc;
    #pragma unroll
    for (int v = 0; v < 8; ++v) acc[v] = 0.f;

    for (int ct = 0; ct < nTiles; ++ct) {
      const int j = ct * 16 + c;
      // branchless B fragment: single ds_load_b64 per pair
      const v2f* bp = (const v2f*)(sB + j * 8);
      v2f b_lo = bp[half];        // half0: (b0,b1)  half1: (b2,b3)
      v2f b_hi = bp[2 + half];    // half0: (b4,b5)  half1: (pad,pad) * A==0

      v8f q;
      #pragma unroll
      for (int v = 0; v < 8; ++v) q[v] = 0.f;
      q = __builtin_amdgcn_wmma_f32_16x16x4_f32(false, a_lo, false, b_lo,
                                                (short)0, q, false, false);
      q = __builtin_amdgcn_wmma_f32_16x16x4_f32(false, a_hi, false, b_hi,
                                                (short)0, q, false, false);
      float wj = sW[j];
      #pragma unroll
      for (int v = 0; v < 8; ++v) acc[v] += wj * __expf(-0.5f * q[v]);
    }
    // reduce over columns: sum across the 16 lanes of each half-group
    #pragma unroll
    for (int v = 0; v < 8; ++v) {
      float s = acc[v];
      for (int off = 1; off < 16; off <<= 1) s += __shfl_xor(s, off, 32);
      if (c == 0) sVals[r0 + v + half * 8] = s;   // D layout: VGPR v -> rows v / v+8
    }
  }
  __syncthreads();

  // ---- Phase 2: bias + ReLU ----
  float bl = bias[l];
  for (int i = tid; i < N; i += 256) sVals[i] = fmaxf(sVals[i] - bl, 0.0f);
  __syncthreads();

  // ---- Phase 3: deterministic top-KSEL (stable: lower index wins ties) ----
  for (int t = 0; t < KSEL; ++t) {
    float bv = -1.0f; int bi = N;
    for (int j = tid; j < N; j += 256) {
      float v = sVals[j];
      if (v > bv) { bv = v; bi = j; }        // ascending scan -> lowest idx on tie
    }
    redV[tid] = bv; redI[tid] = bi;
    __syncthreads();
    for (int s = 128; s > 0; s >>= 1) {
      if (tid < s) {
        float v1 = redV[tid + s]; int i1 = redI[tid + s];
        if (v1 > redV[tid] || (v1 == redV[tid] && i1 < redI[tid])) {
          redV[tid] = v1; redI[tid] = i1;
        }
      }
      __syncthreads();
    }
    if (tid == 0) {
      int sel = redI[0];
      float* o = out + (size_t)((b * L + l) * KSEL + t) * 7;
      const float* g = mb + (size_t)sel * 7;
      o[0] = redV[0];
      o[1] = g[1]; o[2] = g[2]; o[3] = g[3]; o[4] = g[4]; o[5] = g[5]; o[6] = g[6];
      sVals[sel] = -2.0f;                    // exclude from further rounds
    }
    __syncthreads();
  }
}

// ---------------------------------------------------------------------------
// batchnorm (per_gaussian=False): scale[l] = mean_{b,n} |w|*2pi*sqrt(det)
// single block, deterministic tree reductions, scales w in place
// ---------------------------------------------------------------------------
__global__ void __launch_bounds__(256)
bn_kernel(float* m, int B, int L, int N) {
  __shared__ float red[256];
  __shared__ float sScale[16];
  const int tid = threadIdx.x;
  const int cnt = B * N;
  for (int l = 0; l < L; ++l) {
    float v = 0.f;
    for (int e = tid; e < cnt; e += 256) {
      int b = e / N, n = e % N;
      const float* g = m + (size_t)((b * L + l) * N + n) * 7;
      v += fabsf(g[0]) * TWO_PI * sqrtf(det2(g));
    }
    red[tid] = v;
    __syncthreads();
    for (int s = 128; s > 0; s >>= 1) {
      if (tid < s) red[tid] += red[tid + s];
      __syncthreads();
    }
    if (tid == 0) sScale[l] = red[0] / (float)cnt;
    __syncthreads();
  }
  const int total = B * L * N;
  for (int e = tid; e < total; e += 256) {
    int n = e % N; int t = e / N; int l = t % L; int b = t / L;
    float* g = m + (size_t)((b * L + l) * N + n) * 7;
    g[0] = g[0] / (sScale[l] + EPS);
  }
}

// ---------------------------------------------------------------------------
// integrate (4,10,16,7) -> (4,10), then log_softmax over classes
// ---------------------------------------------------------------------------
__global__ void finish_kernel(const float* __restrict__ m, float* __restrict__ out) {
  __shared__ float red[640];
  __shared__ float xs[40];
  const int tid = threadIdx.x;   // 640 threads
  if (tid < 640) {
    const float* g = m + (size_t)tid * 7;
    red[tid] = g[0] * TWO_PI * sqrtf(det2(g));
  }
  __syncthreads();
  for (int off = 8; off > 0; off >>= 1) {
    if (tid < 640 && (tid & 15) < off) red[tid] += red[tid + off];
    __syncthreads();
  }
  if (tid < 640 && (tid & 15) == 0) xs[tid >> 4] = red[tid];
  __syncthreads();
  if (tid < 4) {
    float mx = -1e30f;
    for (int l = 0; l < 10; ++l) mx = fmaxf(mx, xs[tid * 10 + l]);
    float se = 0.f;
    for (int l = 0; l < 10; ++l) se += expf(xs[tid * 10 + l] - mx);
    float lse = logf(se);
    for (int l = 0; l < 10; ++l) out[tid * 10 + l] = xs[tid * 10 + l] - mx - lse;
  }
}

// ---------------------------------------------------------------------------
extern "C" void kernel_launch(void* const* d_in, const int* in_sizes, int n_in,
                              void* d_out, int out_size, void* d_ws, size_t ws_size,
                              hipStream_t stream) {
  const float* in_x = (const float*)d_in[0];
  const float* k1   = (const float*)d_in[1];
  const float* k2   = (const float*)d_in[2];
  const float* k3   = (const float*)d_in[3];
  const float* b1   = (const float*)d_in[4];
  const float* b2   = (const float*)d_in[5];
  const float* b3   = (const float*)d_in[6];
  float* out = (float*)d_out;
  float* ws  = (float*)d_ws;

  float* x0 = ws;               // 4*1*128*7      = 3584
  float* m1 = x0 + 3584;        // 4*8*640*7      = 143360
  float* s1 = m1 + 143360;      // 4*8*32*7       = 7168
  float* m2 = s1 + 7168;        // 4*10*1280*7    = 358400
  float* s2 = m2 + 358400;      // 4*10*16*7      = 4480
  float* m3 = s2 + 4480;        // 4*10*800*7     = 224000
  float* s3 = m3 + 224000;      // 4*10*16*7      = 4480

  bn1_kernel<<<1, 128, 0, stream>>>(in_x, x0);

  conv_kernel<<<(4 * 8 * 640 + 255) / 256, 256, 0, stream>>>(x0, k1, m1, 4, 8, 1, 128, 5);
  bias_relu_kernel<640, 32><<<4 * 8, 256, 0, stream>>>(m1, b1, s1, 8);
  bn_kernel<<<1, 256, 0, stream>>>(s1, 4, 8, 32);

  conv_kernel<<<(4 * 10 * 1280 + 255) / 256, 256, 0, stream>>>(s1, k2, m2, 4, 10, 8, 32, 5);
  bias_relu_kernel<1280, 16><<<4 * 10, 256, 0, stream>>>(m2, b2, s2, 10);
  bn_kernel<<<1, 256, 0, stream>>>(s2, 4, 10, 16);

  conv_kernel<<<(4 * 10 * 800 + 255) / 256, 256, 0, stream>>>(s2, k3, m3, 4, 10, 10, 16, 5);
  bias_relu_kernel<800, 16><<<4 * 10, 256, 0, stream>>>(m3, b3, s3, 10);
  bn_kernel<<<1, 256, 0, stream>>>(s3, 4, 10, 16);

  finish_kernel<<<1, 640, 0, stream>>>(s3, out);
}